// EMD_module_72310069395879
// MI455X (gfx1250) — compile-verified
//
#include <hip/hip_runtime.h>
#include <hip/hip_bf16.h>
#include <math.h>

typedef __attribute__((ext_vector_type(2))) float v2f;
typedef __attribute__((ext_vector_type(8))) float v8f;

#define HW 25
#define CDIM 640
#define CHUNK 32
#define CHF (CHUNK * HW)          /* 800 floats per matrix per chunk */
#define SINK_EPS 0.05f
#define SINK_ITERS 100

__global__ __launch_bounds__(128)
void emd_sinkhorn_kernel(const float* __restrict__ Sg,
                         const float* __restrict__ Qg,
                         float* __restrict__ out) {
    // Chunk tiles kept in EXACT global layout (stride 25, contiguous):
    // loader is a pure float4 memcpy. +2 float4 slack absorbs the harmless
    // WMMA over-read of padded rows/cols (max linear index 806 < 808).
    __shared__ float4 Qs4[CHF / 4 + 2];
    __shared__ float4 Ss4[CHF / 4 + 2];
    __shared__ float GS[32][32];      // Gram matrix Q^T S (later overwritten by sim)
    __shared__ float Km[HW][32];      // Sinkhorn kernel exp(-(1-sim)/eps)
    __shared__ float muQ[HW], nQ[HW], muS[HW], nS[HW];
    __shared__ float av[HW], bvv[HW], uvv[HW], vvv[HW];
    __shared__ float red[128];
    __shared__ float scal[2];

    float* __restrict__ Qs = (float*)Qs4;
    float* __restrict__ Ss = (float*)Ss4;

    const int n    = blockIdx.x;
    const int tid  = threadIdx.x;
    const int lane = tid & 31;
    const int wave = tid >> 5;
    const int m16  = lane & 15;
    const int half = lane >> 4;
    const int mo   = (wave >> 1) << 4;   // tile row offset (0 or 16)
    const int no   = (wave & 1) << 4;    // tile col offset (0 or 16)

    const float* __restrict__ Sb = Sg + (size_t)n * (CDIM * HW);
    const float* __restrict__ Qb = Qg + (size_t)n * (CDIM * HW);

    v8f acc = {};                    // 16x16 fp32 accumulator tile
    float csum = 0.f, csq = 0.f;     // per-column Σx, Σx² (t 0..24: Q, 32..56: S)

    for (int c0 = 0; c0 < CDIM; c0 += CHUNK) {
        __syncthreads();             // previous chunk fully consumed
        {   // vectorized chunk copy: 200 float4 per matrix, zero index math
            const float4* __restrict__ qg = (const float4*)(Qb + c0 * HW);
            const float4* __restrict__ sg = (const float4*)(Sb + c0 * HW);
            for (int i = tid; i < CHF / 4; i += 128) {
                Qs4[i] = qg[i];
                Ss4[i] = sg[i];
            }
            // prefetch next chunk (global_prefetch_b8); 25 cachelines each
            if (c0 + CHUNK < CDIM) {
                if (tid < HW)
                    __builtin_prefetch(Qb + (c0 + CHUNK) * HW + tid * 32, 0, 0);
                else if (tid >= 32 && tid < 32 + HW)
                    __builtin_prefetch(Sb + (c0 + CHUNK) * HW + (tid - 32) * 32, 0, 0);
            }
        }
        __syncthreads();

        // column statistics for centering / norms
        if (tid < HW) {
            for (int r = 0; r < CHUNK; ++r) { float x = Qs[r * HW + tid]; csum += x; csq += x * x; }
        } else if (tid >= 32 && tid < 32 + HW) {
            int t = tid - 32;
            for (int r = 0; r < CHUNK; ++r) { float x = Ss[r * HW + t]; csum += x; csq += x * x; }
        }

        // Gram tile: D = A(16x4) x B(4x16) + C, fp32 exact.
        // Out-of-range cols (>=25) read neighbor-row garbage -> only pollutes
        // D rows/cols >= 25, which are never consumed.
        for (int kb = 0; kb < CHUNK; kb += 4) {
            const int ra = (kb + 2 * half) * HW;
            v2f a, b;
            a.x = Qs[ra + mo + m16];
            a.y = Qs[ra + HW + mo + m16];
            b.x = Ss[ra + no + m16];
            b.y = Ss[ra + HW + no + m16];
            acc = __builtin_amdgcn_wmma_f32_16x16x4_f32(
                false, a, false, b, (short)0, acc, false, false);
        }
    }
    __syncthreads();

    // spill Gram tiles: VGPR r holds M = mo + r + 8*half, N = no + m16
    for (int r = 0; r < 8; ++r) GS[mo + r + 8 * half][no + m16] = acc[r];
    if (tid < HW) {
        float mu = csum / CDIM;
        float s2 = csq - CDIM * mu * mu;
        muQ[tid] = mu;
        nQ[tid]  = fmaxf(sqrtf(fmaxf(s2, 0.f)), 1e-8f);
    } else if (tid >= 32 && tid < 32 + HW) {
        int t = tid - 32;
        float mu = csum / CDIM;
        float s2 = csq - CDIM * mu * mu;
        muS[t] = mu;
        nS[t]  = fmaxf(sqrtf(fmaxf(s2, 0.f)), 1e-8f);
    }
    __syncthreads();

    // marginals from the RAW Gram matrix: w1 = relu(rowmean), w2 = relu(colmean)
    if (tid < HW) {
        float rs = 0.f, cs = 0.f;
        for (int v = 0; v < HW; ++v) { rs += GS[tid][v]; cs += GS[v][tid]; }
        av[tid]  = fmaxf(rs / HW, 0.f) + 0.001f + 1e-5f;
        bvv[tid] = fmaxf(cs / HW, 0.f) + 0.001f + 1e-5f;
    }
    __syncthreads();
    if (tid == 0) {
        float sa = 0.f, sb = 0.f;
        for (int i = 0; i < HW; ++i) { sa += av[i]; sb += bvv[i]; }
        scal[0] = (float)HW / sa; scal[1] = (float)HW / sb;
    }
    __syncthreads();
    if (tid < HW) { av[tid] *= scal[0]; bvv[tid] *= scal[1]; vvv[tid] = 1.f; }

    // centered cosine sim (overwrite GS in place) and Sinkhorn kernel
    for (int idx = tid; idx < HW * HW; idx += 128) {
        int u = idx / HW, v = idx - u * HW;
        float s = (GS[u][v] - (float)CDIM * muQ[u] * muS[v]) / (nQ[u] * nS[v]);
        GS[u][v] = s;
        Km[u][v] = expf(-(1.f - s) / SINK_EPS);
    }
    __syncthreads();

    // Sinkhorn: u = a/(K v), v = b/(K^T u)
    for (int it = 0; it < SINK_ITERS; ++it) {
        if (tid < HW) {
            float s = 0.f;
            for (int j = 0; j < HW; ++j) s += Km[tid][j] * vvv[j];
            uvv[tid] = av[tid] / (s + 1e-30f);
        }
        __syncthreads();
        if (tid < HW) {
            float s = 0.f;
            for (int j = 0; j < HW; ++j) s += Km[j][tid] * uvv[j];
            vvv[tid] = bvv[tid] / (s + 1e-30f);
        }
        __syncthreads();
    }
    if (tid < HW) {
        float s = 0.f;
        for (int j = 0; j < HW; ++j) s += Km[tid][j] * vvv[j];
        uvv[tid] = av[tid] / (s + 1e-30f);
    }
    __syncthreads();

    // logit = sum(sim * flow) * (T/hw), flow = u ⊗ v scaled K
    float part = 0.f;
    for (int idx = tid; idx < HW * HW; idx += 128) {
        int u = idx / HW, v = idx - u * HW;
        part += GS[u][v] * uvv[u] * Km[u][v] * vvv[v];
    }
    red[tid] = part;
    __syncthreads();
    for (int s = 64; s > 0; s >>= 1) {
        if (tid < s) red[tid] += red[tid + s];
        __syncthreads();
    }
    if (tid == 0) out[n] = red[0] * (12.5f / (float)HW);
}

extern "C" void kernel_launch(void* const* d_in, const int* in_sizes, int n_in,
                              void* d_out, int out_size, void* d_ws, size_t ws_size,
                              hipStream_t stream) {
    const float* S = (const float*)d_in[0];   // in_support
    const float* Q = (const float*)d_in[1];   // in_query
    float* out = (float*)d_out;
    int N = in_sizes[0] / (CDIM * HW);        // 600
    emd_sinkhorn_kernel<<<N, 128, 0, stream>>>(S, Q, out);
}